// RecurrentLayer_83339545411621
// MI455X (gfx1250) — compile-verified
//
#include <hip/hip_runtime.h>

// Problem dims (fixed by reference)
#define BDIM 64
#define TDIM 512
#define IDIM 128
#define HDIM 512

#define NWG        16          // persistent workgroups, one per WGP
#define CH_PER_WG  32          // HDIM / NWG output channels per WG
#define THREADS    256         // 8 wave32 per WG -> 4 (M) x 2 (N) tiles of 16x16

// LDS row strides (float units)
#define WH_STRIDE 516          // W_hid slice rows: 16 lanes -> 16 bank groups
#define WI_STRIDE 132
#define XT_STRIDE 132
#define FR_STRIDE 132          // fr chunks: TDM pads each 128-DWORD row by 4 DWORDs

#define KCHUNK       128                      // fr staged as 4 chunks of 128 K-values
#define FR_CHUNK_FL  (BDIM * FR_STRIDE)       // floats per staged fr chunk (64x132)

typedef __attribute__((ext_vector_type(2))) float    v2f;
typedef __attribute__((ext_vector_type(8))) float    v8f;
typedef __attribute__((ext_vector_type(4))) unsigned v4u;
typedef __attribute__((ext_vector_type(8))) int      v8i;
typedef __attribute__((ext_vector_type(4))) int      v4i;

// ---- CDNA5 async / TDM helpers ------------------------------------------
__device__ __forceinline__ unsigned ldsAddr(const void* p) {
  return (unsigned)(__SIZE_TYPE__)(__attribute__((address_space(3))) const void*)p;
}
// GLOBAL_LOAD_ASYNC_TO_LDS_B128 (ASYNCcnt): copies 16 bytes from the global base in saddr plus the offset in vaddr to the LDS address in vdst
__device__ __forceinline__ void asyncF4(unsigned lds_byte, unsigned long long gbase,
                                        unsigned gbyte) {
  asm volatile("global_load_async_to_lds_b128 %0, %1, %2"
               :: "v"(lds_byte), "v"(gbyte), "s"(gbase) : "memory");
}
__device__ __forceinline__ void waitAsync0() {
  asm volatile("s_wait_asynccnt 0" ::: "memory");
}
__device__ __forceinline__ void waitTensor(int n) {  // n is compile-time after unroll
  switch (n) {
    case 0:  __builtin_amdgcn_s_wait_tensorcnt(0); break;
    case 1:  __builtin_amdgcn_s_wait_tensorcnt(1); break;
    case 2:  __builtin_amdgcn_s_wait_tensorcnt(2); break;
    default: __builtin_amdgcn_s_wait_tensorcnt(3); break;
  }
}

// TENSOR_LOAD_TO_LDS of one fr chunk: 2D tile 128(x) x 64(y) fp32, row stride 512,
// LDS padding 4 DWORDs per 128-DWORD row -> dense 64x132-float chunk in LDS.
__device__ __forceinline__ void tdm_load_fr_chunk(const float* rb, float* frS, int c) {
  unsigned long long ga = (unsigned long long)(rb + c * KCHUNK);
  unsigned lds = ldsAddr(frS + c * FR_CHUNK_FL);
  v4u g0 = { 1u,                                          // count=1, user descriptor
             lds,                                         // lds_addr
             (unsigned)ga,                                // global_addr[31:0]
             ((unsigned)(ga >> 32) & 0x01FFFFFFu) | (2u << 30) };  // ga[56:32] | type=2
  v8i g1 = { (int)((2u << 16)      // data_size = 4B
                 | (1u << 20)      // pad_enable
                 | (6u << 22)      // pad_interval: 128 DWORDs
                 | (3u << 25)),    // pad_amount: 4 DWORDs
             (int)(128u << 16),    // tensor_dim0 = 128
             (int)(64u  << 16),    // tensor_dim1 = 64
             (int)(128u << 16),    // tile_dim0 = 128
             64,                   // tile_dim1 = 64 (tile_dim2 = 0)
             512,                  // tensor_dim0_stride = 512 elements
             0, 0 };               // tensor_dim1_stride unused (2D)
  v4i z4 = { 0, 0, 0, 0 };
  v8i z8 = { 0, 0, 0, 0, 0, 0, 0, 0 };
  __builtin_amdgcn_tensor_load_to_lds(g0, g1, z4, z4, z8, 0);  // 6-arg (clang-23 form)
}

// ---- split grid barrier (persistent-kernel scan sync) -------------------
__device__ __forceinline__ void grid_arrive(int* __restrict__ ctr, int idx) {
  __threadfence();          // every thread's fr/out stores reach device scope
  __syncthreads();
  if (threadIdx.x == 0)
    __hip_atomic_fetch_add(&ctr[idx], 1, __ATOMIC_RELEASE, __HIP_MEMORY_SCOPE_AGENT);
}
__device__ __forceinline__ void grid_wait(int* __restrict__ ctr, int idx) {
  if (threadIdx.x == 0) {
    while (__hip_atomic_load(&ctr[idx], __ATOMIC_ACQUIRE, __HIP_MEMORY_SCOPE_AGENT) < NWG)
      __builtin_amdgcn_s_sleep(1);
  }
  __syncthreads();
}

// x tile prefetch on the ASYNCcnt path (independent of the recurrence)
__device__ __forceinline__ void prefetch_x(const float* x, int t, float* xtb, int tid) {
  for (int i = tid; i < BDIM * (IDIM / 4); i += THREADS) {   // 8 B128 per wave
    int r = i >> 5, c4 = i & 31;
    unsigned gb = (unsigned)(((size_t)r * TDIM * IDIM + (size_t)t * IDIM + c4 * 4) * 4);
    asyncF4(ldsAddr(xtb + r * XT_STRIDE + c4 * 4), (unsigned long long)x, gb);
  }
}

extern "C" __global__ __launch_bounds__(THREADS, 1)
void rnn_scan_kernel(const float* __restrict__ x,          // [B,T,I]
                     const float* __restrict__ init_state, // [B,H]
                     const float* __restrict__ W_in,       // [H,I]
                     const float* __restrict__ b_in,       // [H]
                     const float* __restrict__ W_hid,      // [H,H]
                     const float* __restrict__ b_hid,      // [H]
                     const float* __restrict__ alpha,      // [H]
                     float* __restrict__ out,              // states ++ relax, each [B,T,H]
                     float* __restrict__ frbuf,            // ws: 2 * B*H ping-pong fr state
                     int* __restrict__ ctr)                // ws: T+2 counters (zeroed)
{
  extern __shared__ float smem[];
  float* frS = smem;                               // 4 chunks of [64][132] (TDM dest)
  float* whS = frS + 4 * FR_CHUNK_FL;              // [32][516] W_hid slice (resident)
  float* wiS = whS + CH_PER_WG * WH_STRIDE;        // [32][132] W_in slice  (resident)
  float* xt0 = wiS + CH_PER_WG * WI_STRIDE;        // [64][132] x tile, double buffered
  float* xt1 = xt0 + BDIM * XT_STRIDE;

  const int tid  = threadIdx.x;
  const int wg   = blockIdx.x;
  const int wave = tid >> 5;
  const int lane = tid & 31;
  const int lo   = lane & 15;
  const int hi   = lane >> 4;
  const int m0   = (wave & 3) * 16;                // batch tile base
  const int n0   = (wave >> 2) * 16;               // local channel tile base
  const int chg  = wg * CH_PER_WG + n0 + lo;       // global output channel (N=lo)

  // ---- one-time: weight slices -> LDS (reused for all 512 steps) ----
  {
    const float4* wh4 = (const float4*)(W_hid + (size_t)wg * CH_PER_WG * HDIM);
    for (int i = tid; i < CH_PER_WG * (HDIM / 4); i += THREADS) {
      int r = i / (HDIM / 4), c4 = i % (HDIM / 4);
      *(float4*)(whS + r * WH_STRIDE + c4 * 4) = wh4[i];
    }
    const float4* wi4 = (const float4*)(W_in + (size_t)wg * CH_PER_WG * IDIM);
    for (int i = tid; i < CH_PER_WG * (IDIM / 4); i += THREADS) {
      int r = i / (IDIM / 4), c4 = i % (IDIM / 4);
      *(float4*)(wiS + r * WI_STRIDE + c4 * 4) = wi4[i];
    }
  }

  // ---- per-lane persistent recurrent state ----
  const float bh = b_hid[chg];
  const float bi = b_in[chg];
  const float al = alpha[chg];
  float vst[8];
  float* buf0 = frbuf;                 // read at even t
  float* buf1 = frbuf + BDIM * HDIM;   // read at odd t
#pragma unroll
  for (int r = 0; r < 8; ++r) {
    int b = m0 + hi * 8 + r;           // C/D layout: VGPR r -> M = r + 8*hi
    float v0 = init_state[b * HDIM + chg];
    vst[r] = v0;
    buf0[b * HDIM + chg] = fmaxf(v0, 0.0f);   // fr0 = relu(init_state)
  }

  prefetch_x(x, 0, xt0, tid);          // x tile for t=0, overlaps barrier 0
  grid_arrive(ctr, 0);

  // Per-lane LDS operand bases (A 16x4: [M=lo][K=2*hi..]; B 4x16: [K=2*hi..][N=lo])
  const int aBaseH = (m0 + lo) * FR_STRIDE + 2 * hi;   // within one fr chunk
  const int bBaseH = (n0 + lo) * WH_STRIDE + 2 * hi;
  const int aBaseI = (m0 + lo) * XT_STRIDE + 2 * hi;
  const int bBaseI = (n0 + lo) * WI_STRIDE + 2 * hi;

  for (int t = 0; t < TDIM; ++t) {
    float* xcur = (t & 1) ? xt1 : xt0;
    float* xnxt = (t & 1) ? xt0 : xt1;
    const float* rb = (t & 1) ? buf1 : buf0;
    float*       wb = (t & 1) ? buf0 : buf1;

    // x_t tile resident for the whole WG (covers t=0 and the t-1 prefetch)
    waitAsync0();
    __syncthreads();

    // ---- input GEMM first: independent of the recurrence, overlaps
    //      other WGs still finishing step t-1 ----
    v8f accI0 = {0.f,0.f,0.f,0.f,0.f,0.f,0.f,0.f};
    v8f accI1 = {0.f,0.f,0.f,0.f,0.f,0.f,0.f,0.f};
#pragma unroll 8
    for (int k = 0; k < IDIM; k += 8) {
      v2f a0 = *(const v2f*)(xcur + aBaseI + k);
      v2f b0 = *(const v2f*)(wiS  + bBaseI + k);
      accI0 = __builtin_amdgcn_wmma_f32_16x16x4_f32(false, a0, false, b0,
                                                    (short)0, accI0, false, false);
      v2f a1 = *(const v2f*)(xcur + aBaseI + k + 4);
      v2f b1 = *(const v2f*)(wiS  + bBaseI + k + 4);
      accI1 = __builtin_amdgcn_wmma_f32_16x16x4_f32(false, a1, false, b1,
                                                    (short)0, accI1, false, false);
    }

    // ---- wait for fr(t-1), then TDM-stage it in 4 chunks on TENSORcnt,
    //      computing chunk c while chunks c+1.. stream via the DMA engine ----
    grid_wait(ctr, t);

    if (wave == 0) {                    // one wave drives the Tensor Data Mover
      tdm_load_fr_chunk(rb, frS, 0);
      tdm_load_fr_chunk(rb, frS, 1);
      tdm_load_fr_chunk(rb, frS, 2);
      tdm_load_fr_chunk(rb, frS, 3);
    }
    if (t + 1 < TDIM) prefetch_x(x, t + 1, xnxt, tid);   // overlaps hidden GEMM

    v8f accH0 = {0.f,0.f,0.f,0.f,0.f,0.f,0.f,0.f};
    v8f accH1 = {0.f,0.f,0.f,0.f,0.f,0.f,0.f,0.f};
#pragma unroll
    for (int c = 0; c < 4; ++c) {
      if (wave == 0) waitTensor(3 - c);   // in-order retire: chunk c landed
      __syncthreads();
      const int aC = c * FR_CHUNK_FL + aBaseH;
      const int bC = c * KCHUNK + bBaseH;
#pragma unroll 8
      for (int k = 0; k < KCHUNK; k += 8) {
        v2f a0 = *(const v2f*)(frS + aC + k);
        v2f b0 = *(const v2f*)(whS + bC + k);
        accH0 = __builtin_amdgcn_wmma_f32_16x16x4_f32(false, a0, false, b0,
                                                      (short)0, accH0, false, false);
        v2f a1 = *(const v2f*)(frS + aC + k + 4);
        v2f b1 = *(const v2f*)(whS + bC + k + 4);
        accH1 = __builtin_amdgcn_wmma_f32_16x16x4_f32(false, a1, false, b1,
                                                      (short)0, accH1, false, false);
      }
    }

    // ---- leaky-integrator epilogue + ReLU ----
#pragma unroll
    for (int r = 0; r < 8; ++r) {
      int b = m0 + hi * 8 + r;
      float vh   = (accH0[r] + accH1[r]) + bh;
      float vi   = (accI0[r] + accI1[r]) + bi;
      float vnew = (1.0f - al) * vst[r] + al * (vh + vi);
      float vr   = (1.0f - al) * vnew + al * vh;
      float fn   = fmaxf(vnew, 0.0f);
      float frx  = fmaxf(vr, 0.0f);
      vst[r] = vnew;
      wb[b * HDIM + chg] = fn;
      size_t o = (size_t)b * TDIM * HDIM + (size_t)t * HDIM + chg;
      out[o] = fn;
      out[(size_t)BDIM * TDIM * HDIM + o] = frx;
    }

    grid_arrive(ctr, t + 1);   // arrive only; next step waits after its input GEMM
  }
}

extern "C" void kernel_launch(void* const* d_in, const int* in_sizes, int n_in,
                              void* d_out, int out_size, void* d_ws, size_t ws_size,
                              hipStream_t stream) {
  (void)in_sizes; (void)n_in; (void)out_size; (void)ws_size;
  const float* x          = (const float*)d_in[0];
  const float* init_state = (const float*)d_in[1];
  const float* W_in       = (const float*)d_in[2];
  const float* b_in       = (const float*)d_in[3];
  const float* W_hid      = (const float*)d_in[4];
  const float* b_hid      = (const float*)d_in[5];
  const float* alpha      = (const float*)d_in[6];

  float* frbuf = (float*)d_ws;                                   // 2 * B*H floats
  int*   ctr   = (int*)((char*)d_ws + (size_t)2 * BDIM * HDIM * sizeof(float));

  // capture-legal: reset barrier counters every launch (ws is not re-poisoned)
  (void)hipMemsetAsync(ctr, 0, (TDIM + 2) * sizeof(int), stream);

  size_t shmem = (size_t)(4 * FR_CHUNK_FL + CH_PER_WG * WH_STRIDE +
                          CH_PER_WG * WI_STRIDE + 2 * BDIM * XT_STRIDE) * sizeof(float);
  // 285696 B dynamic LDS (<= 320 KB WGP LDS); raise the per-kernel cap
  (void)hipFuncSetAttribute((const void*)rnn_scan_kernel,
                            hipFuncAttributeMaxDynamicSharedMemorySize, (int)shmem);

  rnn_scan_kernel<<<NWG, THREADS, shmem, stream>>>(
      x, init_state, W_in, b_in, W_hid, b_hid, alpha,
      (float*)d_out, frbuf, ctr);
}